// Ascend310Qwen3Next_GatedDeltaNet_7249904795769
// MI455X (gfx1250) — compile-verified
//
#include <hip/hip_runtime.h>
#include <math.h>

// ---------------- problem constants ----------------
#define HIDDEN     2048
#define NUM_KH     16
#define NUM_VH     32
#define HEAD_K     128
#define HEAD_V     128
#define NTOK       4096
#define QKVZ_DIM   12288     // 16*128*2 + 32*128*2
#define BA_DIM     64        // 2*32
#define KHEAD_STR  768       // per-k-head stride inside qkvz row
#define EPS        1e-6f

typedef __attribute__((ext_vector_type(2))) float v2f;
typedef __attribute__((ext_vector_type(4))) float v4f;
typedef __attribute__((ext_vector_type(8))) float v8f;

// ---- CDNA5 async global->LDS copy (ASYNCcnt path), with fallback ----
#if defined(__has_builtin)
#  if __has_builtin(__builtin_amdgcn_global_load_async_to_lds_b128)
#    define USE_ASYNC 1
#  endif
#  if __has_builtin(__builtin_amdgcn_s_wait_asynccnt)
#    define WAIT_ASYNC0() __builtin_amdgcn_s_wait_asynccnt(0)
#  endif
#endif
#ifndef USE_ASYNC
#  define USE_ASYNC 0
#endif
#ifndef WAIT_ASYNC0
#  define WAIT_ASYNC0() asm volatile("s_wait_asynccnt 0" ::: "memory")
#endif

// builtin signature (from the round-2 diagnostic): param0 is
// "__attribute__((__vector_size__(4*sizeof(int)))) int __device__ *"
typedef int v4i __attribute__((vector_size(16)));
typedef __attribute__((address_space(1))) v4i* gas_v4p;
typedef __attribute__((address_space(3))) v4i* las_v4p;

// =====================================================================
// FP32 WMMA GEMM:  C[M,N] = A[M,K] @ B[K,N], row-major fp32.
// 64x128 block tile, 256 threads = 8 waves (2x4), 2x2 WMMA accs/wave.
// Double-buffered LDS stages fed by async global->LDS b128 copies.
// =====================================================================
#define BM 64
#define BN 128
#define BK 16
#define ASTR 20    // A tile [m][k] row stride: 16B-aligned rows, conflict-free
#define BSTR 144   // B tile [k][n] row stride: rows k and k+2 hit disjoint banks

__global__ __launch_bounds__(256) void gemm_f32_wmma(
    const float* __restrict__ A, const float* __restrict__ B,
    float* __restrict__ C, int M, int N, int K)
{
    __shared__ float As[2][BM * ASTR];
    __shared__ float Bs[2][BK * BSTR];

    const int tid   = threadIdx.x;
    const int lane  = tid & 31;
    const int wid   = tid >> 5;       // 0..7
    const int wm    = wid >> 2;       // 0..1
    const int wn    = wid & 3;        // 0..3
    const int lhalf = lane >> 4;      // K-pair select (A/B), +8 rows (C)
    const int l16   = lane & 15;

    const int block_m = blockIdx.y * BM;
    const int block_n = blockIdx.x * BN;

    // per-thread copy coordinates (16B chunks)
    const int am  = tid >> 2;                // A row 0..63
    const int ac  = (tid & 3) << 2;          // A col chunk
    const int bk0 = tid >> 5;                // B row 0..7 (and +8)
    const int bc0 = (tid & 31) << 2;         // B col chunk

    auto copy_stage = [&](int buf, int k0) {
        const float* ga  = A + (size_t)(block_m + am) * K + k0 + ac;
        const float* gb0 = B + (size_t)(k0 + bk0) * N + block_n + bc0;
        const float* gb1 = B + (size_t)(k0 + bk0 + 8) * N + block_n + bc0;
        float* la  = &As[buf][am * ASTR + ac];
        float* lb0 = &Bs[buf][bk0 * BSTR + bc0];
        float* lb1 = &Bs[buf][(bk0 + 8) * BSTR + bc0];
#if USE_ASYNC
        __builtin_amdgcn_global_load_async_to_lds_b128((gas_v4p)ga,  (las_v4p)la,  0, 0);
        __builtin_amdgcn_global_load_async_to_lds_b128((gas_v4p)gb0, (las_v4p)lb0, 0, 0);
        __builtin_amdgcn_global_load_async_to_lds_b128((gas_v4p)gb1, (las_v4p)lb1, 0, 0);
#else
        *(v4f*)la  = *(const v4f*)ga;
        *(v4f*)lb0 = *(const v4f*)gb0;
        *(v4f*)lb1 = *(const v4f*)gb1;
#endif
    };

    v8f acc[2][2] = {};

    copy_stage(0, 0);
    const int nstages = K / BK;
    for (int s = 0; s < nstages; ++s) {
#if USE_ASYNC
        WAIT_ASYNC0();
#endif
        __syncthreads();
        if (s + 1 < nstages) copy_stage((s + 1) & 1, (s + 1) * BK);

        const float* as_ = As[s & 1];
        const float* bs_ = Bs[s & 1];

        // batch all fragment loads for the stage, then a clean WMMA chain
        v2f af[4][2], bf[4][2];
        #pragma unroll
        for (int ks4 = 0; ks4 < 4; ++ks4) {
            const int kk = ks4 * 4 + (lhalf << 1);
            #pragma unroll
            for (int i = 0; i < 2; ++i) {
                const int m = wm * 32 + i * 16 + l16;
                af[ks4][i] = *(const v2f*)&as_[m * ASTR + kk];
            }
            #pragma unroll
            for (int j = 0; j < 2; ++j) {
                const int n = wn * 32 + j * 16 + l16;
                bf[ks4][j].x = bs_[kk * BSTR + n];
                bf[ks4][j].y = bs_[(kk + 1) * BSTR + n];
            }
        }
        #pragma unroll
        for (int ks4 = 0; ks4 < 4; ++ks4)
            #pragma unroll
            for (int i = 0; i < 2; ++i)
                #pragma unroll
                for (int j = 0; j < 2; ++j)
                    acc[i][j] = __builtin_amdgcn_wmma_f32_16x16x4_f32(
                        false, af[ks4][i], false, bf[ks4][j],
                        (short)0, acc[i][j], false, false);
    }

    // C layout: VGPR g -> M=g (lanes 0-15) / M=g+8 (lanes 16-31), N = l16
    #pragma unroll
    for (int i = 0; i < 2; ++i)
        #pragma unroll
        for (int j = 0; j < 2; ++j) {
            const int rbase = block_m + wm * 32 + i * 16 + (lhalf << 3);
            const int col   = block_n + wn * 32 + j * 16 + l16;
            #pragma unroll
            for (int g = 0; g < 8; ++g)
                C[(size_t)(rbase + g) * N + col] = acc[i][j][g];
        }
}

// =====================================================================
// Tiny GEMM for ba: one block per token row; H row staged in LDS.
// =====================================================================
__global__ __launch_bounds__(256) void gemm_ba_kernel(
    const float* __restrict__ H, const float* __restrict__ Wba,
    float* __restrict__ BA)
{
    __shared__ float hrow[HIDDEN];
    __shared__ float partial[256];
    const int tid = threadIdx.x;
    const int t   = blockIdx.x;
    for (int i = tid; i < HIDDEN; i += 256)
        hrow[i] = H[(size_t)t * HIDDEN + i];
    __syncthreads();
    const int n = tid & 63;
    const int c = tid >> 6;
    float s = 0.f;
    #pragma unroll 4
    for (int k = c * 512; k < c * 512 + 512; ++k)
        s = fmaf(hrow[k], Wba[(size_t)k * BA_DIM + n], s);
    partial[tid] = s;
    __syncthreads();
    if (tid < 64)
        BA[(size_t)t * BA_DIM + tid] =
            partial[tid] + partial[tid + 64] + partial[tid + 128] + partial[tid + 192];
}

// =====================================================================
// Sequential delta-rule scan, 2-way K-split.
// One block per v-head (32 blocks), 256 threads: thread (kc, v) owns
// state rows S[kc*64 .. kc*64+63][v] in registers (64 VGPRs).
// Per-token critical path: ~192 FMA + 2 LDS partial-sum exchanges.
// Prefetches next token's q/k/v/z lines (global_prefetch_b8).
// =====================================================================
__device__ __forceinline__ float wave_sum32(float x) {
    #pragma unroll
    for (int off = 16; off > 0; off >>= 1)
        x += __shfl_xor(x, off, 32);
    return x;
}

__global__ __launch_bounds__(256) void deltanet_scan(
    const float* __restrict__ qkvz, const float* __restrict__ ba,
    const float* __restrict__ A_log, const float* __restrict__ dt_bias,
    const float* __restrict__ norm_w, float* __restrict__ X)
{
    const int h   = blockIdx.x;       // v-head 0..31
    const int tid = threadIdx.x;      // 0..255
    const int v   = tid & 127;
    const int kc  = tid >> 7;         // 0/1: K half
    const int kh  = h >> 1;
    const int r   = h & 1;
    const int lane = tid & 31, w = tid >> 5;
    const int kbase = kc * 64;

    __shared__ float qs[HEAD_K], ks[HEAD_K];
    __shared__ float redq[8], redk[8], redo[8];
    __shared__ float kvpart[256], outpart[256];

    float S[64];
    #pragma unroll
    for (int i = 0; i < 64; ++i) S[i] = 0.f;

    const float negA  = -expf(A_log[h]);
    const float dtb   = dt_bias[h];
    const float nw    = norm_w[v];
    const float qscal = rsqrtf((float)HEAD_K);

    for (int t = 0; t < NTOK; ++t) {
        const float* base = qkvz + (size_t)t * QKVZ_DIM + kh * KHEAD_STR;
        const float qv  = base[v];
        const float kvl = base[HEAD_K + v];
        const float vt  = base[2 * HEAD_K + r * HEAD_V + v];
        const float zt  = base[2 * HEAD_K + 2 * HEAD_V + r * HEAD_V + v];

        // prefetch next token's lines (one byte/lane -> whole lines)
        if (kc == 0 && t + 1 < NTOK) {
            const float* nb = base + QKVZ_DIM;
            __builtin_prefetch(nb + v, 0, 1);                                  // q
            __builtin_prefetch(nb + HEAD_K + v, 0, 1);                         // k
            __builtin_prefetch(nb + 2 * HEAD_K + r * HEAD_V + v, 0, 1);        // v
            __builtin_prefetch(nb + 2 * HEAD_K + 2 * HEAD_V + r * HEAD_V + v, 0, 1); // z
        }

        // ---- l2 norms of q,k (each value counted twice -> *0.5) ----
        float pq = wave_sum32(qv * qv);
        float pk = wave_sum32(kvl * kvl);
        if (lane == 0) { redq[w] = pq; redk[w] = pk; }
        __syncthreads();                                   // B1
        float sq = 0.f, sk = 0.f;
        #pragma unroll
        for (int i = 0; i < 8; ++i) { sq += redq[i]; sk += redk[i]; }
        sq *= 0.5f; sk *= 0.5f;
        if (kc == 0) {
            qs[v] = qv  * rsqrtf(sq + EPS) * qscal;
            ks[v] = kvl * rsqrtf(sk + EPS);
        }
        __syncthreads();                                   // B2

        // ---- gating scalars (uniform across block) ----
        const float b    = ba[(size_t)t * BA_DIM + kh * 4 + r];
        const float a    = ba[(size_t)t * BA_DIM + kh * 4 + 2 + r];
        const float beta = 1.f / (1.f + expf(-b));
        float sp = a + dtb;
        sp = (sp > 20.f) ? sp : log1pf(expf(sp));
        const float eg = expf(negA * sp);

        // ---- kv partial over this thread's 64 K rows ----
        float kvp = 0.f;
        #pragma unroll
        for (int i = 0; i < 64; ++i) kvp = fmaf(ks[kbase + i], S[i], kvp);
        kvpart[tid] = kvp;
        __syncthreads();                                   // B3
        const float kv    = (kvpart[v] + kvpart[128 + v]) * eg;
        const float delta = (vt - kv) * beta;

        // ---- S = S*e^g + k*delta ; out partial = q . S_new ----
        float op = 0.f;
        #pragma unroll
        for (int i = 0; i < 64; ++i) {
            S[i] = fmaf(ks[kbase + i], delta, S[i] * eg);
            op   = fmaf(qs[kbase + i], S[i], op);
        }
        outpart[tid] = op;
        __syncthreads();                                   // B4
        const float out = outpart[v] + outpart[128 + v];

        // ---- RMSNorm over 128 v-lanes + SiLU(z) gate ----
        float po = wave_sum32(out * out);
        if (lane == 0) redo[w] = po;
        __syncthreads();                                   // B5
        float so = 0.f;
        #pragma unroll
        for (int i = 0; i < 8; ++i) so += redo[i];
        const float mean = so * 0.5f * (1.f / (float)HEAD_V);
        if (kc == 0) {
            float x = out * rsqrtf(mean + EPS) * nw;
            x *= zt / (1.f + expf(-zt));
            X[(size_t)t * (NUM_VH * HEAD_V) + h * HEAD_V + v] = x;
        }
    }
}

// =====================================================================
// launcher
// =====================================================================
extern "C" void kernel_launch(void* const* d_in, const int* in_sizes, int n_in,
                              void* d_out, int out_size, void* d_ws, size_t ws_size,
                              hipStream_t stream)
{
    const float* H      = (const float*)d_in[0];
    const float* Wqkvz  = (const float*)d_in[1];
    const float* Wba    = (const float*)d_in[2];
    const float* A_log  = (const float*)d_in[3];
    const float* dtb    = (const float*)d_in[4];
    const float* normw  = (const float*)d_in[5];
    const float* Wout   = (const float*)d_in[6];
    float* out = (float*)d_out;

    char* ws = (char*)d_ws;
    const size_t QKVZ_BYTES = (size_t)NTOK * QKVZ_DIM * sizeof(float); // 192 MB
    const size_t BA_BYTES   = (size_t)NTOK * BA_DIM * sizeof(float);   // 1 MB
    float* qkvz = (float*)ws;
    float* ba   = (float*)(ws + QKVZ_BYTES);
    float* X    = (float*)(ws + QKVZ_BYTES + BA_BYTES);                // 64 MB

    // 1) qkvz = H @ W_qkvz   (4096 x 12288, K=2048)
    gemm_f32_wmma<<<dim3(QKVZ_DIM / BN, NTOK / BM), 256, 0, stream>>>(
        H, Wqkvz, qkvz, NTOK, QKVZ_DIM, HIDDEN);

    // 2) ba = H @ W_ba       (4096 x 64)
    gemm_ba_kernel<<<NTOK, 256, 0, stream>>>(H, Wba, ba);

    // 3) sequential scan + norms + gating -> X (4096 x 4096)
    deltanet_scan<<<NUM_VH, 256, 0, stream>>>(qkvz, ba, A_log, dtb, normw, X);

    // 4) out = X @ W_out     (4096 x 2048, K=4096)
    gemm_f32_wmma<<<dim3(HIDDEN / BN, NTOK / BM), 256, 0, stream>>>(
        X, Wout, out, NTOK, HIDDEN, NUM_VH * HEAD_V);
}